// MACE_88502096101411
// MI455X (gfx1250) — compile-verified
//
#include <hip/hip_runtime.h>
#include <hip/hip_bf16.h>
#include <math.h>

// ---------------- problem constants (match reference) ----------------
#define NN      4096      // N_NODES
#define NE      65536     // N_EDGES
#define NG      128       // N_GRAPHS
#define CD      128       // C
#define NELEM   100
#define INV_AVG (1.0f/35.0f)
#define INV_SQRT_ELEM 0.1f     // 1/sqrt(100)

typedef __attribute__((ext_vector_type(2))) float v2f;
typedef __attribute__((ext_vector_type(8))) float v8f;

__device__ __forceinline__ float silu_f(float x) { return x / (1.0f + __expf(-x)); }

// =====================================================================
// fp32 WMMA GEMM, register-blocked 2x2 per wave:  D = act( alpha*A@B + Cadd )
// - Block: 128 threads = 4 waves arranged 2x2 -> 64x64 output tile.
// - Wave: 32x32 output (4 accumulators); per k-step 2 A-frags + 2 B-frags
//   feed 4 x V_WMMA_F32_16X16X4_F32 -> 1 LDS + 1 global operand per WMMA.
// - A rows remapped: memRow(r) = (r/rpn)*strideR + baseR + (r%rpn), which
//   implements einsum('n m c, c d -> n m d') over (node,m) row pairs.
// Requires M % 64 == 0, N % 64 == 0, K % 4 == 0 (true for all call sites).
// =====================================================================
__global__ void k_gemm_wmma(const float* __restrict__ A, int ldA, int rpn, int strA, int basA,
                            const float* __restrict__ B, int ldB,
                            float* __restrict__ D, int ldD, int strD, int basD,
                            const float* __restrict__ Cadd, int ldC,
                            int K, float alpha, int act)
{
    extern __shared__ float ldsA[];       // 64 x K panel
    const int tid  = threadIdx.x;         // 0..127
    const int wave = tid >> 5;            // 0..3
    const int wr   = wave >> 1;           // wave row   (0..1) -> 32 rows
    const int wc   = wave & 1;            // wave col   (0..1) -> 32 cols
    const int lane = tid & 31;
    const int half = lane >> 4;           // 0: lanes 0-15, 1: lanes 16-31
    const int l16  = lane & 15;
    const int rowBase = blockIdx.x * 64;
    const int colBase = blockIdx.y * 64;

    // cooperatively stage the 64 x K A panel (row-remapped) into LDS
    for (int idx = tid; idx < 64 * K; idx += 128) {
        int rr = idx / K, kk = idx - rr * K;
        int r  = rowBase + rr;
        int mr = (r / rpn) * strA + basA + (r % rpn);
        ldsA[rr * K + kk] = A[(size_t)mr * ldA + kk];
    }
    __syncthreads();

    v8f acc00 = {0.f,0.f,0.f,0.f,0.f,0.f,0.f,0.f};
    v8f acc01 = acc00, acc10 = acc00, acc11 = acc00;

    const int ar0  = wr * 32 + l16;       // LDS rows for the two A fragments
    const int ar1  = ar0 + 16;
    const int col0 = colBase + wc * 32 + l16;   // global cols for the two B frags
    const int col1 = col0 + 16;

    for (int k0 = 0; k0 < K; k0 += 4) {
        // A 16x4 frag: lanes 0-15 K=k0,k0+1 ; lanes 16-31 K=k0+2,k0+3
        // B 4x16 frag mirrors the half-lane K split.
        int ka = k0 + half * 2;
        v2f a0, a1, b0, b1;
        a0.x = ldsA[ar0 * K + ka];  a0.y = ldsA[ar0 * K + ka + 1];
        a1.x = ldsA[ar1 * K + ka];  a1.y = ldsA[ar1 * K + ka + 1];
        b0.x = B[(size_t)ka * ldB + col0];  b0.y = B[(size_t)(ka + 1) * ldB + col0];
        b1.x = B[(size_t)ka * ldB + col1];  b1.y = B[(size_t)(ka + 1) * ldB + col1];
        acc00 = __builtin_amdgcn_wmma_f32_16x16x4_f32(false, a0, false, b0, (short)0, acc00, false, false);
        acc01 = __builtin_amdgcn_wmma_f32_16x16x4_f32(false, a0, false, b1, (short)0, acc01, false, false);
        acc10 = __builtin_amdgcn_wmma_f32_16x16x4_f32(false, a1, false, b0, (short)0, acc10, false, false);
        acc11 = __builtin_amdgcn_wmma_f32_16x16x4_f32(false, a1, false, b1, (short)0, acc11, false, false);
    }

    // D 16x16 tile layout: lanes 0-15 -> rows i, lanes 16-31 -> rows i+8
    auto store_tile = [&](v8f acc, int tr, int tc) {
#pragma unroll
        for (int i = 0; i < 8; ++i) {
            int r  = rowBase + wr * 32 + tr * 16 + i + half * 8;
            int mr = (r / rpn) * strD + basD + (r % rpn);
            int cc = colBase + wc * 32 + tc * 16 + l16;
            float v = alpha * acc[i];
            if (Cadd) v += Cadd[(size_t)mr * ldC + cc];
            if (act)  v = silu_f(v);
            D[(size_t)mr * ldD + cc] = v;
        }
    };
    store_tile(acc00, 0, 0);
    store_tile(acc01, 0, 1);
    store_tile(acc10, 1, 0);
    store_tile(acc11, 1, 1);
}

static inline void launch_gemm(hipStream_t st,
                               const float* A, int ldA, int rpn, int strA, int basA,
                               const float* B, int ldB,
                               float* D, int ldD, int strD, int basD,
                               const float* Cadd, int ldC,
                               int M, int N, int K, float alpha, int act)
{
    dim3 grid(M / 64, N / 64);
    size_t lds = (size_t)64 * K * sizeof(float);   // <= 32 KB (K <= 128)
    k_gemm_wmma<<<grid, 128, lds, st>>>(A, ldA, rpn, strA, basA, B, ldB,
                                        D, ldD, strD, basD, Cadd, ldC, K, alpha, act);
}

// ---------------- small helper kernels ----------------
__global__ void k_zero(float* p, int n) {
    int i = blockIdx.x * blockDim.x + threadIdx.x;
    if (i < n) p[i] = 0.f;
}

// edge geometry: Yb[e][0..3] = {1, sqrt3*uy, sqrt3*uz, sqrt3*ux}; ef[e][0..7] bessel*cutoff
__global__ void k_edge_geom(const float* __restrict__ ea, float* __restrict__ Yb,
                            float* __restrict__ ef)
{
    int e = blockIdx.x * blockDim.x + threadIdx.x;
    if (e >= NE) return;
    float vx = ea[e * 3 + 0], vy = ea[e * 3 + 1], vz = ea[e * 3 + 2];
    float r  = sqrtf(vx * vx + vy * vy + vz * vz);
    float inv = 1.0f / r;
    float ux = vx * inv, uy = vy * inv, uz = vz * inv;
    const float SQ3 = 1.7320508075688772f;
    Yb[e * 4 + 0] = 1.0f;
    Yb[e * 4 + 1] = SQ3 * uy;
    Yb[e * 4 + 2] = SQ3 * uz;
    Yb[e * 4 + 3] = SQ3 * ux;
    float x  = r * 0.2f;                       // r / R_MAX
    float x2 = x * x, x5 = x2 * x2 * x, x6 = x5 * x, x7 = x6 * x;
    float cut = (x < 1.0f) ? (1.0f - 21.0f * x5 + 35.0f * x6 - 15.0f * x7) : 0.0f;
    const float KSTEP = 0.6283185307179586f;   // pi / R_MAX
    const float AMP   = 0.6324555320336759f;   // sqrt(2/R_MAX)
#pragma unroll
    for (int k = 0; k < 8; ++k)
        ef[e * 8 + k] = AMP * __sinf(r * KSTEP * (float)(k + 1)) * inv * cut;
}

// h[n,c] = W_emb[Z[n], c]   (one_hot @ W_emb == row gather)
__global__ void k_embed(const int* __restrict__ Z, const float* __restrict__ W_emb,
                        float* __restrict__ h)
{
    int n = blockIdx.x, c = threadIdx.x;
    h[(size_t)n * CD + c] = W_emb[(size_t)Z[n] * CD + c];
}

// layer-0 message scatter: only lm 0..3 survive pruning (A0[2],A0[3] are dead)
__global__ void k_scatter0(const float* __restrict__ Rsel, const float* __restrict__ x0up,
                           const float* __restrict__ Yb, const int* __restrict__ snd,
                           const int* __restrict__ rcv, float* __restrict__ A0sum)
{
    int e = blockIdx.x, c = threadIdx.x;
    int s = snd[e], r = rcv[e];
    float xs = x0up[(size_t)s * CD + c];
    float y1 = Yb[e * 4 + 1], y2 = Yb[e * 4 + 2], y3 = Yb[e * 4 + 3];
    float m0 = Rsel[(size_t)e * 256 + c] * xs;          // l=0 radial channel
    float m1 = Rsel[(size_t)e * 256 + 128 + c] * xs;    // l=1 radial channel
    float* base = A0sum + ((size_t)r * 4) * CD + c;
    atomicAdd(base + 0 * CD, m0);
    atomicAdd(base + 1 * CD, m1 * y1);
    atomicAdd(base + 2 * CD, m1 * y2);
    atomicAdd(base + 3 * CD, m1 * y3);
}

// layer-1 message scatter: only msgs[0] survives (A1[1..3] dead).
// msg0 = R1[:,0,:]*x0s + R1[:,5,:] * sum_a x1s[:,:,a] * Y1[a] * (1/sqrt(3))
// (CG(1,1,0)[a,b,0] = delta_ab / sqrt(3), derived analytically from _real_cg)
__global__ void k_scatter1(const float* __restrict__ Rsel, const float* __restrict__ x0,
                           const float* __restrict__ x1, const float* __restrict__ Yb,
                           const int* __restrict__ snd, const int* __restrict__ rcv,
                           float* __restrict__ A1sum)
{
    int e = blockIdx.x, c = threadIdx.x;
    int s = snd[e], r = rcv[e];
    const float IS3 = 0.57735026918962576f;
    float g0 = Yb[e * 4 + 1] * IS3, g1 = Yb[e * 4 + 2] * IS3, g2 = Yb[e * 4 + 3] * IS3;
    size_t b1 = ((size_t)s * 3) * CD + c;
    float dot = x1[b1] * g0 + x1[b1 + CD] * g1 + x1[b1 + 2 * CD] * g2;
    float val = Rsel[(size_t)e * 256 + c] * x0[(size_t)s * CD + c]
              + Rsel[(size_t)e * 256 + 128 + c] * dot;
    atomicAdd(&A1sum[(size_t)r * CD + c], val);
}

// per-node element-specific skip: out[n,lm,d] = 0.1 * sum_c s2[n,lm,c]*W_skip0[l,Z,c,d]
__global__ void k_skip0(const float* __restrict__ s2, const float* __restrict__ Wsk,
                        const int* __restrict__ Z, float* __restrict__ out)
{
    __shared__ float row[CD];
    int n = blockIdx.x, d = threadIdx.x, z = Z[n];
    for (int lm = 0; lm < 4; ++lm) {
        int l = (lm > 0) ? 1 : 0;
        row[d] = s2[((size_t)n * 4 + lm) * CD + d];
        __syncthreads();
        const float* W = Wsk + ((size_t)(l * NELEM + z)) * CD * CD;
        float acc = 0.f;
#pragma unroll 4
        for (int c = 0; c < CD; ++c) acc += row[c] * W[c * CD + d];
        out[((size_t)n * 4 + lm) * CD + d] = acc * INV_SQRT_ELEM;
        __syncthreads();
    }
}

// sc[n,d] = 0.1 * sum_c h0[n,c] * W_sc1[Z[n],c,d]
__global__ void k_sc1(const float* __restrict__ h0, const float* __restrict__ Wsc,
                      const int* __restrict__ Z, float* __restrict__ sc)
{
    __shared__ float row[CD];
    int n = blockIdx.x, d = threadIdx.x, z = Z[n];
    row[d] = h0[(size_t)n * CD + d];
    __syncthreads();
    const float* W = Wsc + (size_t)z * CD * CD;
    float acc = 0.f;
#pragma unroll 4
    for (int c = 0; c < CD; ++c) acc += row[c] * W[c * CD + d];
    sc[(size_t)n * CD + d] = acc * INV_SQRT_ELEM;
}

// polynomial gate layer 0: A0p[n,lm,c] = A0sk[n,lm,c] * poly_l(sA[n,c])
__global__ void k_poly0(const float* __restrict__ A0sk, const float* __restrict__ Wp,
                        const int* __restrict__ Z, float* __restrict__ A0p)
{
    int n = blockIdx.x, c = threadIdx.x, z = Z[n];
    float sA  = A0sk[((size_t)n * 4) * CD + c];
    float sA2 = sA * sA;
    {   // l = 0
        float w0 = Wp[((size_t)(0 * 3 + 0) * NELEM + z) * CD + c];
        float w1 = Wp[((size_t)(0 * 3 + 1) * NELEM + z) * CD + c];
        float w2 = Wp[((size_t)(0 * 3 + 2) * NELEM + z) * CD + c];
        float p  = w0 + w1 * sA + w2 * sA2;
        A0p[((size_t)n * 4 + 0) * CD + c] = sA * p;
    }
    {   // l = 1 (same gate for all 3 m-components)
        float w0 = Wp[((size_t)(1 * 3 + 0) * NELEM + z) * CD + c];
        float w1 = Wp[((size_t)(1 * 3 + 1) * NELEM + z) * CD + c];
        float w2 = Wp[((size_t)(1 * 3 + 2) * NELEM + z) * CD + c];
        float p  = w0 + w1 * sA + w2 * sA2;
#pragma unroll
        for (int a = 0; a < 3; ++a)
            A0p[((size_t)n * 4 + 1 + a) * CD + c] =
                A0sk[((size_t)n * 4 + 1 + a) * CD + c] * p;
    }
}

// polynomial gate layer 1: hpre = sA * (w0 + w1*sA + w2*sA^2)
__global__ void k_poly1(const float* __restrict__ sA1, const float* __restrict__ Wp,
                        const int* __restrict__ Z, float* __restrict__ hpre)
{
    int n = blockIdx.x, c = threadIdx.x, z = Z[n];
    float sA = sA1[(size_t)n * CD + c];
    float w0 = Wp[((size_t)0 * NELEM + z) * CD + c];
    float w1 = Wp[((size_t)1 * NELEM + z) * CD + c];
    float w2 = Wp[((size_t)2 * NELEM + z) * CD + c];
    hpre[(size_t)n * CD + c] = sA * (w0 + w1 * sA + w2 * sA * sA);
}

// node readout (C=128 -> 9) + per-graph accumulation
__global__ void k_readout(const float* __restrict__ hout, const float* __restrict__ Wr,
                          const int* __restrict__ batch, float* __restrict__ gsum,
                          float* __restrict__ gcnt)
{
    int n = blockIdx.x * blockDim.x + threadIdx.x;
    if (n >= NN) return;
    float acc[9];
#pragma unroll
    for (int d = 0; d < 9; ++d) acc[d] = 0.f;
    for (int c = 0; c < CD; ++c) {
        float hv = hout[(size_t)n * CD + c];
#pragma unroll
        for (int d = 0; d < 9; ++d) acc[d] += hv * Wr[c * 9 + d];
    }
    int g = batch[n];
#pragma unroll
    for (int d = 0; d < 9; ++d) atomicAdd(&gsum[g * 9 + d], acc[d]);
    atomicAdd(&gcnt[g], 1.0f);
}

__global__ void k_final(const float* __restrict__ gsum, const float* __restrict__ gcnt,
                        float* __restrict__ out)
{
    int i = blockIdx.x * blockDim.x + threadIdx.x;
    if (i >= NG * 9) return;
    out[i] = gsum[i] / fmaxf(gcnt[i / 9], 1.0f);
}

// =====================================================================
extern "C" void kernel_launch(void* const* d_in, const int* in_sizes, int n_in,
                              void* d_out, int out_size, void* d_ws, size_t ws_size,
                              hipStream_t stream)
{
    (void)in_sizes; (void)n_in; (void)out_size; (void)ws_size;
    const int*   Z         = (const int*)d_in[0];
    const float* edge_attr = (const float*)d_in[1];
    const int*   eidx      = (const int*)d_in[2];
    const int*   snd       = eidx;
    const int*   rcv       = eidx + NE;
    const int*   batch     = (const int*)d_in[3];
    const float* W_emb     = (const float*)d_in[4];
    const float* W_up0     = (const float*)d_in[5];
    const float* r0w1      = (const float*)d_in[6];
    const float* r0w2      = (const float*)d_in[7];
    const float* r0w3      = (const float*)d_in[8];
    const float* r0w4      = (const float*)d_in[9];
    const float* W_lin0    = (const float*)d_in[10];
    const float* W_skip0   = (const float*)d_in[11];
    const float* Wp0_scale = (const float*)d_in[12];
    const float* Wp0_mix   = (const float*)d_in[13];
    const float* W_sc1     = (const float*)d_in[14];
    const float* Wup1_0    = (const float*)d_in[15];
    const float* Wup1_1    = (const float*)d_in[16];
    const float* r1w1      = (const float*)d_in[17];
    const float* r1w2      = (const float*)d_in[18];
    const float* r1w3      = (const float*)d_in[19];
    const float* r1w4      = (const float*)d_in[20];
    const float* W_lin1    = (const float*)d_in[21];
    const float* Wp1_scale = (const float*)d_in[22];
    const float* Wp1_mix   = (const float*)d_in[23];
    const float* W_read    = (const float*)d_in[24];
    float* out = (float*)d_out;

    // -------- workspace carve-up (floats) --------
    float* ws = (float*)d_ws;
    size_t off = 0;
    auto alloc = [&](size_t nf) { float* p = ws + off; off += nf; return p; };
    float* h     = alloc((size_t)NN * CD);
    float* x0up  = alloc((size_t)NN * CD);
    float* Yb    = alloc((size_t)NE * 4);
    float* ef    = alloc((size_t)NE * 8);
    float* t1    = alloc((size_t)NE * 64);
    float* t2    = alloc((size_t)NE * 64);
    float* Rsel  = alloc((size_t)NE * 256);   // reused for R0 (l=0,1) then R1 (p=0,5)
    float* A0sum = alloc((size_t)NN * 4 * CD);
    float* s2    = alloc((size_t)NN * 4 * CD);
    float* A0sk  = alloc((size_t)NN * 4 * CD);
    float* A0p   = alloc((size_t)NN * 4 * CD);
    float* h0    = alloc((size_t)NN * CD);
    float* h1    = alloc((size_t)NN * 3 * CD);
    float* scb   = alloc((size_t)NN * CD);
    float* x0    = alloc((size_t)NN * CD);
    float* x1    = alloc((size_t)NN * 3 * CD);
    float* A1sum = alloc((size_t)NN * CD);
    float* sA1   = alloc((size_t)NN * CD);
    float* hpre  = alloc((size_t)NN * CD);
    float* hout  = alloc((size_t)NN * CD);
    float* gsum  = alloc((size_t)NG * 9);
    float* gcnt  = alloc((size_t)NG);         // contiguous after gsum (zeroed together)

    // -------- stage 1: geometry + embedding + up-projection --------
    k_edge_geom<<<NE / 256, 256, 0, stream>>>(edge_attr, Yb, ef);
    k_embed<<<NN, CD, 0, stream>>>(Z, W_emb, h);
    launch_gemm(stream, h, CD, 1, 1, 0, W_up0, CD, x0up, CD, 1, 0, nullptr, 0,
                NN, CD, CD, 1.f, 0);

    // -------- stage 2: R0 MLP (only l=0,1 output columns needed) --------
    launch_gemm(stream, ef, 8, 1, 1, 0, r0w1, 64, t1, 64, 1, 0, nullptr, 0, NE, 64, 8, 1.f, 1);
    launch_gemm(stream, t1, 64, 1, 1, 0, r0w2, 64, t2, 64, 1, 0, nullptr, 0, NE, 64, 64, 1.f, 1);
    launch_gemm(stream, t2, 64, 1, 1, 0, r0w3, 64, t1, 64, 1, 0, nullptr, 0, NE, 64, 64, 1.f, 1);
    launch_gemm(stream, t1, 64, 1, 1, 0, r0w4, 512, Rsel, 256, 1, 0, nullptr, 0, NE, 256, 64, 1.f, 0);

    // -------- stage 3: message 0 segment-sum (fp32 atomics into L2) --------
    k_zero<<<(NN * 4 * CD + 255) / 256, 256, 0, stream>>>(A0sum, NN * 4 * CD);
    k_scatter0<<<NE, CD, 0, stream>>>(Rsel, x0up, Yb, snd, rcv, A0sum);

    // -------- stage 4: W_lin0 (alpha = 1/35 folds segment-mean) --------
    launch_gemm(stream, A0sum, CD, 1, 4, 0, W_lin0, CD, s2, CD, 4, 0, nullptr, 0,
                NN, CD, CD, INV_AVG, 0);                               // l=0
    launch_gemm(stream, A0sum, CD, 3, 4, 1, W_lin0 + CD * CD, CD, s2, CD, 4, 1, nullptr, 0,
                3 * NN, CD, CD, INV_AVG, 0);                           // l=1 (3 m rows)

    // -------- stage 5: element-specific skip, gate, mix --------
    k_skip0<<<NN, CD, 0, stream>>>(s2, W_skip0, Z, A0sk);
    k_poly0<<<NN, CD, 0, stream>>>(A0sk, Wp0_scale, Z, A0p);
    launch_gemm(stream, A0p, CD, 1, 4, 0, Wp0_mix, CD, h0, CD, 1, 0, nullptr, 0,
                NN, CD, CD, 1.f, 0);                                   // feats l=0 -> h0
    launch_gemm(stream, A0p, CD, 3, 4, 1, Wp0_mix + CD * CD, CD, h1, CD, 3, 0, nullptr, 0,
                3 * NN, CD, CD, 1.f, 0);                               // feats l=1 -> h1

    // -------- stage 6: layer-1 node projections --------
    k_sc1<<<NN, CD, 0, stream>>>(h0, W_sc1, Z, scb);
    launch_gemm(stream, h0, CD, 1, 1, 0, Wup1_0, CD, x0, CD, 1, 0, nullptr, 0,
                NN, CD, CD, 1.f, 0);
    launch_gemm(stream, h1, CD, 3, 3, 0, Wup1_1, CD, x1, CD, 3, 0, nullptr, 0,
                3 * NN, CD, CD, 1.f, 0);

    // -------- stage 7: R1 MLP (only channels p=0 and p=5 survive) --------
    launch_gemm(stream, ef, 8, 1, 1, 0, r1w1, 64, t1, 64, 1, 0, nullptr, 0, NE, 64, 8, 1.f, 1);
    launch_gemm(stream, t1, 64, 1, 1, 0, r1w2, 64, t2, 64, 1, 0, nullptr, 0, NE, 64, 64, 1.f, 1);
    launch_gemm(stream, t2, 64, 1, 1, 0, r1w3, 64, t1, 64, 1, 0, nullptr, 0, NE, 64, 64, 1.f, 1);
    launch_gemm(stream, t1, 64, 1, 1, 0, r1w4, 1280, Rsel, 256, 1, 0, nullptr, 0,
                NE, 128, 64, 1.f, 0);                                  // cols 0..127   (p=0)
    launch_gemm(stream, t1, 64, 1, 1, 0, r1w4 + 640, 1280, Rsel + 128, 256, 1, 0, nullptr, 0,
                NE, 128, 64, 1.f, 0);                                  // cols 640..767 (p=5)

    // -------- stage 8: message 1 segment-sum (single channel after pruning) --------
    k_zero<<<(NN * CD + 255) / 256, 256, 0, stream>>>(A1sum, NN * CD);
    k_scatter1<<<NE, CD, 0, stream>>>(Rsel, x0, x1, Yb, snd, rcv, A1sum);

    // -------- stage 9: W_lin1[0], gate, mix (+sc), readout --------
    launch_gemm(stream, A1sum, CD, 1, 1, 0, W_lin1, CD, sA1, CD, 1, 0, nullptr, 0,
                NN, CD, CD, INV_AVG, 0);
    k_poly1<<<NN, CD, 0, stream>>>(sA1, Wp1_scale, Z, hpre);
    launch_gemm(stream, hpre, CD, 1, 1, 0, Wp1_mix, CD, hout, CD, 1, 0, scb, CD,
                NN, CD, CD, 1.f, 0);                                   // + sc addend

    k_zero<<<(NG * 9 + NG + 255) / 256, 256, 0, stream>>>(gsum, NG * 9 + NG);
    k_readout<<<NN / 64, 64, 0, stream>>>(hout, W_read, batch, gsum, gcnt);
    k_final<<<(NG * 9 + 127) / 128, 128, 0, stream>>>(gsum, gcnt, out);
}